// GraphConv_47467978555683
// MI455X (gfx1250) — compile-verified
//
#include <hip/hip_runtime.h>
#include <stdint.h>

typedef float v2f __attribute__((ext_vector_type(2)));
typedef float v4f __attribute__((ext_vector_type(4)));
typedef float v8f __attribute__((ext_vector_type(8)));

#define NFEAT 256
#define N_NODES 100000
#define LDSA_STRIDE 260   // 16 rows, stride 260 floats -> bank = (4*row + k) % 64, conflict-free

// Guaranteed hardware f32 atomic adds (no CAS-loop fallback), 4 consecutive floats.
__device__ __forceinline__ void atomic_add_f32x4(float* p, float v0, float v1,
                                                 float v2, float v3) {
    asm volatile(
        "global_atomic_add_f32 %0, %1, off\n\t"
        "global_atomic_add_f32 %0, %2, off offset:4\n\t"
        "global_atomic_add_f32 %0, %3, off offset:8\n\t"
        "global_atomic_add_f32 %0, %4, off offset:12"
        :: "v"(p), "v"(v0), "v"(v1), "v"(v2), "v"(v3)
        : "memory");
}

// ---------------- Kernel 1: pair-packed transpose of weight ----------------
// wtp[k2][n] = float2( weight[n][2*k2], weight[n][2*k2+1] )
// i.e. a 256(n) x 128(k2) -> 128(k2) x 256(n) transpose over float2 elements.
__global__ void __launch_bounds__(256)
transpose_w_kernel(const float2* __restrict__ w2, float2* __restrict__ wtp) {
    __shared__ float2 tile[32][33];
    const int bx = blockIdx.x & 3;      // k2 tile (128/32)
    const int by = blockIdx.x >> 2;     // n tile  (256/32)
    const int tx = threadIdx.x & 31;
    const int ty = threadIdx.x >> 5;    // 0..7
    #pragma unroll
    for (int i = 0; i < 32; i += 8)
        tile[ty + i][tx] = w2[(size_t)(by * 32 + ty + i) * 128 + bx * 32 + tx];
    __syncthreads();
    #pragma unroll
    for (int i = 0; i < 32; i += 8)
        wtp[(size_t)(bx * 32 + ty + i) * NFEAT + by * 32 + tx] = tile[tx][ty + i];
}

// ---------------- Kernel 2: y = x @ weight^T via V_WMMA_F32_16X16X4_F32 ----------------
// Block = 256 threads (8 waves). Block computes a 16-row M stripe x all 256 N cols.
// Each wave owns two 16-wide N tiles.
__global__ void __launch_bounds__(256)
gemm_xwt_kernel(const float* __restrict__ x, const float2* __restrict__ wtp,
                float* __restrict__ y) {
    __shared__ float a_lds[16 * LDSA_STRIDE];

    const int tid = threadIdx.x;
    const int m0  = blockIdx.x * 16;

    // Stage 16x256 A tile into LDS with 16B coalesced loads (NT: x is read once).
    #pragma unroll
    for (int i = 0; i < 4; ++i) {
        const int flat = tid + i * 256;       // float4 index, 0..1023
        const int row  = flat >> 6;           // 64 float4s per row
        const int c4   = flat & 63;
        const v4f v = __builtin_nontemporal_load(
            ((const v4f*)(x + (size_t)(m0 + row) * NFEAT)) + c4);
        *((v4f*)&a_lds[row * LDSA_STRIDE + c4 * 4]) = v;
    }
    __syncthreads();

    const int lane    = tid & 31;
    const int wv      = tid >> 5;             // wave id 0..7
    const int halfsel = lane >> 4;            // 0: lanes 0-15, 1: lanes 16-31
    const int lrow    = lane & 15;            // row (A) / col (B,C) within tile
    const int koff    = halfsel * 2;          // ISA layout: K={0,1} vs K={2,3}
    const int n0      = wv * 32;

    v8f acc0 = {};
    v8f acc1 = {};
    const float* arow = &a_lds[lrow * LDSA_STRIDE];

    #pragma unroll 4
    for (int k0 = 0; k0 < NFEAT; k0 += 4) {
        // A fragment: 16x4 f32 -> 2 VGPRs/lane (from LDS, bank-conflict-free)
        v2f a;
        a.x = arow[k0 + koff];
        a.y = arow[k0 + koff + 1];

        // B fragments: single b64 load each from pair-packed transposed weight
        const float2* bp = wtp + (size_t)((k0 + koff) >> 1) * NFEAT + n0 + lrow;
        const float2 t0 = bp[0];
        const float2 t1 = bp[16];
        v2f b0; b0.x = t0.x; b0.y = t0.y;
        v2f b1; b1.x = t1.x; b1.y = t1.y;

        acc0 = __builtin_amdgcn_wmma_f32_16x16x4_f32(false, a, false, b0,
                                                     (short)0, acc0, false, false);
        acc1 = __builtin_amdgcn_wmma_f32_16x16x4_f32(false, a, false, b1,
                                                     (short)0, acc1, false, false);
    }

    // C/D layout: VGPR r -> M = r + 8*halfsel, N = lrow
    #pragma unroll
    for (int i = 0; i < 8; ++i) {
        const int row = i + halfsel * 8;
        y[(size_t)(m0 + row) * NFEAT + n0 + lrow]      = acc0[i];
        y[(size_t)(m0 + row) * NFEAT + n0 + 16 + lrow] = acc1[i];
    }
}

// ---------------- Kernel 3: out[dst] += val * y[src]  (scatter over edges) ----------------
// 64 threads per edge (float4 per thread over 256 feats), 4 edges per 256-thread block.
// Edge lists are streamed once -> non-temporal so they don't evict y/out from L2.
__global__ void __launch_bounds__(256)
scatter_edges_kernel(const float* __restrict__ y,
                     const long long* __restrict__ esrc,
                     const long long* __restrict__ edst,
                     const float* __restrict__ eval,
                     float* __restrict__ out, int nedge) {
    const int tid    = threadIdx.x;
    const int eLocal = tid >> 6;          // 0..3
    const int f4     = tid & 63;          // float4 slot within 256 feats
    const long long e = (long long)blockIdx.x * 4 + eLocal;
    if (e >= nedge) return;

    const long long s = __builtin_nontemporal_load(esrc + e);
    const long long d = __builtin_nontemporal_load(edst + e);
    const float     v = __builtin_nontemporal_load(eval + e);

    const v4f xv = ((const v4f*)(y + (size_t)s * NFEAT))[f4];
    float* op = out + (size_t)d * NFEAT + (size_t)f4 * 4;

    atomic_add_f32x4(op, v * xv.x, v * xv.y, v * xv.z, v * xv.w);
}

extern "C" void kernel_launch(void* const* d_in, const int* in_sizes, int n_in,
                              void* d_out, int out_size, void* d_ws, size_t ws_size,
                              hipStream_t stream) {
    const float*     x    = (const float*)d_in[0];
    const long long* esrc = (const long long*)d_in[1];   // int64 per reference
    const long long* edst = (const long long*)d_in[2];   // int64 per reference
    const float*     eval = (const float*)d_in[3];
    const float*     w    = (const float*)d_in[4];
    float*           out  = (float*)d_out;

    const int nedge = in_sizes[1];

    // workspace: pair-packed wT (256KB) then y (100000*256 f32 = 102.4MB)
    float2* wtp = (float2*)d_ws;
    float*  y   = (float*)((char*)d_ws + (size_t)NFEAT * NFEAT * sizeof(float));

    // 1) pair-packed weight transpose
    transpose_w_kernel<<<32, 256, 0, stream>>>((const float2*)w, wtp);

    // 2) y = x @ W^T (WMMA f32)
    gemm_xwt_kernel<<<N_NODES / 16, 256, 0, stream>>>(x, wtp, y);

    // 3) zero output, then scatter-add val * y[src] into out[dst]
    (void)hipMemsetAsync(d_out, 0, (size_t)out_size * sizeof(float), stream);
    scatter_edges_kernel<<<(nedge + 3) / 4, 256, 0, stream>>>(y, esrc, edst, eval, out, nedge);
}